// Block_9345848836513
// MI455X (gfx1250) — compile-verified
//
#include <hip/hip_runtime.h>
#include <math.h>

// Problem constants (match reference)
#define BB 4
#define CC 32
#define TT 131072
#define WINDOW 512
#define HOP 256
#define NCO 257          // WINDOW/2 + 1
#define NF (TT / HOP)    // 512 frames
#define SROW 514         // spec row: 257 re + 257 im floats

typedef float v2f __attribute__((ext_vector_type(2)));
typedef float v8f __attribute__((ext_vector_type(8)));

// ---------------------------------------------------------------------------
// Stage A: channel mix  xm[b,d,t] = sum_c x[b,c,t] * M[c,d]
// GEMM per batch: D(32 x T) = A(32x32=M^T) * B(32 x T), via WMMA f32 16x16x4.
// One wave per 16(d) x 16(t) output tile, K=32 as 8 chained k=4 WMMAs.
// ---------------------------------------------------------------------------
__global__ __launch_bounds__(256) void mix_wmma(const float* __restrict__ x,
                                                const float* __restrict__ mixer,
                                                float* __restrict__ xm) {
  const int lane = threadIdx.x & 31;
  const int wave = (blockIdx.x * (blockDim.x >> 5)) + (threadIdx.x >> 5);
  const int tTiles = TT / 16;                 // 8192
  const int ttile = wave % tTiles;
  const int dtile = (wave / tTiles) & 1;      // 0 or 1
  const int b     = wave / (tTiles * 2);      // 0..3

  const int n    = lane & 15;                 // column within tile (t)
  const int half = lane >> 4;                 // lane-half selects K offset
  const int d0   = dtile * 16;
  const int t0   = ttile * 16;

  v8f acc = {0.f, 0.f, 0.f, 0.f, 0.f, 0.f, 0.f, 0.f};

#pragma unroll
  for (int kc = 0; kc < 8; ++kc) {
    const int c0 = kc * 4;
    const int k0 = half * 2;                  // lanes 16-31 hold K=2,3
    // A fragment (16x4): A[m][k] = M[c0+k][d0+m]
    v2f a;
    a.x = mixer[(c0 + k0)     * CC + d0 + n];
    a.y = mixer[(c0 + k0 + 1) * CC + d0 + n];
    // B fragment (4x16): B[k][n] = x[b, c0+k, t0+n]
    const float* xb = x + ((size_t)(b * CC + c0 + k0)) * TT + t0 + n;
    v2f bb;
    bb.x = xb[0];
    bb.y = xb[TT];
    acc = __builtin_amdgcn_wmma_f32_16x16x4_f32(
        /*neg_a=*/false, a, /*neg_b=*/false, bb,
        /*c_mod=*/(short)0, acc, /*reuse_a=*/false, /*reuse_b=*/false);
  }

  // D 16x16 f32 layout: vgpr i holds row m = i + 8*half, lanes = columns
#pragma unroll
  for (int i = 0; i < 8; ++i) {
    const int m = i + 8 * half;
    xm[((size_t)(b * CC + d0 + m)) * TT + t0 + n] = acc[i];
  }
}

__device__ __forceinline__ int bitrev9(int j) {
  return (int)(__brev((unsigned)j) >> 23);
}

// ---------------------------------------------------------------------------
// Stage B: forward 512-pt FFT per frame (real input), write bins 0..256
// One block (256 thr) per frame. spec layout: [frameRow][k] re, [257+k] im.
// ---------------------------------------------------------------------------
__global__ __launch_bounds__(256) void fft_fwd(const float* __restrict__ xm,
                                               float* __restrict__ spec) {
  __shared__ float re[WINDOW];
  __shared__ float im[WINDOW];
  const int frame = blockIdx.x;               // (b*C + c)*NF + f
  const int f     = frame % NF;
  const int bc    = frame / NF;
  const int tid   = threadIdx.x;
  const size_t xoff = (size_t)bc * TT + (size_t)f * HOP;

#pragma unroll
  for (int h = 0; h < 2; ++h) {
    const int j = tid + h * 256;
    const long t = (long)f * HOP + j;
    const float v = (t < TT) ? xm[xoff + j] : 0.f;   // zero pad (last frame)
    const int r = bitrev9(j);
    re[r] = v;
    im[r] = 0.f;
  }

  for (int s = 1; s <= 9; ++s) {
    __syncthreads();
    const int m    = 1 << s;
    const int hlf  = m >> 1;
    const int j    = tid & (hlf - 1);
    const int i0   = ((tid >> (s - 1)) << s) + j;
    const int i1   = i0 + hlf;
    const float ang = -6.28318530717958647692f * (float)j / (float)m;
    const float wr = __cosf(ang), wi = __sinf(ang);
    const float vr = re[i1], vi = im[i1];
    const float tr = wr * vr - wi * vi;
    const float ti = wr * vi + wi * vr;
    const float ur = re[i0], ui = im[i0];
    re[i0] = ur + tr;  im[i0] = ui + ti;
    re[i1] = ur - tr;  im[i1] = ui - ti;
  }
  __syncthreads();

  float* row = spec + (size_t)frame * SROW;
  row[tid]       = re[tid];
  row[257 + tid] = im[tid];
  if (tid == 0) {                            // Nyquist bin
    row[256]       = re[256];
    row[257 + 256] = im[256];
  }
}

// ---------------------------------------------------------------------------
// Stage C: per-(b,c,k) complex scan across frames: cur = (s + prev) * tr
// ---------------------------------------------------------------------------
__global__ __launch_bounds__(256) void scan_frames(float* __restrict__ spec,
                                                   const float* __restrict__ transfer) {
  const int idx = blockIdx.x * blockDim.x + threadIdx.x;
  const int total = BB * CC * NCO;
  if (idx >= total) return;
  const int k = idx % NCO;
  const int c = (idx / NCO) % CC;
  const int b = idx / (NCO * CC);
  const float tr = transfer[c * NCO + k];
  const size_t base = ((size_t)(b * CC + c)) * NF * SROW + k;

  float pr = 0.f, pi = 0.f;
  for (int f = 0; f < NF; ++f) {
    const size_t rb = base + (size_t)f * SROW;
    if (f + 8 < NF) __builtin_prefetch(spec + rb + 8 * SROW, 0, 0);
    const float cr = (spec[rb]       + pr) * tr;
    const float ci = (spec[rb + 257] + pi) * tr;
    spec[rb]       = cr;
    spec[rb + 257] = ci;
    pr = cr;
    pi = ci;
  }
}

// ---------------------------------------------------------------------------
// Stage D: inverse FFT per frame + Hann + overlap-add into out.
// parity 0: even frames tile [0,T) disjointly -> plain store.
// parity 1: odd frames disjoint among themselves -> read-modify-write add.
// ---------------------------------------------------------------------------
__global__ __launch_bounds__(256) void ifft_frame(const float* __restrict__ spec,
                                                  float* __restrict__ out,
                                                  int parity) {
  __shared__ float re[WINDOW];
  __shared__ float im[WINDOW];
  const int fi  = blockIdx.x % (NF / 2);
  const int bc  = blockIdx.x / (NF / 2);
  const int f   = 2 * fi + parity;
  const int tid = threadIdx.x;
  const float* row = spec + ((size_t)(bc * NF + f)) * SROW;

#pragma unroll
  for (int h = 0; h < 2; ++h) {
    const int j = tid + h * 256;
    float xr, xi;
    if (j <= 256) {                       // stored half
      xr = row[j];
      xi = row[257 + j];
    } else {                              // conjugate symmetry
      xr =  row[512 - j];
      xi = -row[257 + (512 - j)];
    }
    const int r = bitrev9(j);
    re[r] = xr;
    im[r] = xi;
  }

  for (int s = 1; s <= 9; ++s) {
    __syncthreads();
    const int m   = 1 << s;
    const int hlf = m >> 1;
    const int j   = tid & (hlf - 1);
    const int i0  = ((tid >> (s - 1)) << s) + j;
    const int i1  = i0 + hlf;
    const float ang = 6.28318530717958647692f * (float)j / (float)m;  // +i: inverse
    const float wr = __cosf(ang), wi = __sinf(ang);
    const float vr = re[i1], vi = im[i1];
    const float tr = wr * vr - wi * vi;
    const float ti = wr * vi + wi * vr;
    const float ur = re[i0], ui = im[i0];
    re[i0] = ur + tr;  im[i0] = ui + ti;
    re[i1] = ur - tr;  im[i1] = ui - ti;
  }
  __syncthreads();

  const size_t ooff = (size_t)bc * TT;
#pragma unroll
  for (int h = 0; h < 2; ++h) {
    const int n = tid + h * 256;
    const long t = (long)f * HOP + n;
    if (t < TT) {
      const float hann = 0.5f - 0.5f * __cosf(6.28318530717958647692f * (float)n / (float)WINDOW);
      const float val = re[n] * (1.0f / (float)WINDOW) * hann;
      if (parity == 0) out[ooff + t] = val;
      else             out[ooff + t] += val;
    }
  }
}

// ---------------------------------------------------------------------------
// Stage E: out = tanh(out * gain[c])
// ---------------------------------------------------------------------------
__global__ __launch_bounds__(256) void tanh_gain(float* __restrict__ out,
                                                 const float* __restrict__ gain) {
  const size_t i = (size_t)blockIdx.x * blockDim.x + threadIdx.x;
  if (i >= (size_t)BB * CC * TT) return;
  const int c = (int)((i / TT) % CC);
  out[i] = tanhf(out[i] * gain[c]);
}

// ---------------------------------------------------------------------------
extern "C" void kernel_launch(void* const* d_in, const int* in_sizes, int n_in,
                              void* d_out, int out_size, void* d_ws, size_t ws_size,
                              hipStream_t stream) {
  (void)in_sizes; (void)n_in; (void)out_size; (void)ws_size;
  const float* x        = (const float*)d_in[0];  // (B,C,T)
  const float* transfer = (const float*)d_in[1];  // (C,257)
  const float* mixer    = (const float*)d_in[2];  // (C,C)
  const float* gain     = (const float*)d_in[3];  // (C,)
  float* out = (float*)d_out;                     // (B,C,T)

  float* xm   = (float*)d_ws;                               // B*C*T floats
  float* spec = xm + (size_t)BB * CC * TT;                  // B*C*NF*SROW floats

  // A: channel mix (WMMA): B*2*(T/16) wave-tiles, 8 waves/block
  {
    const int waves = BB * 2 * (TT / 16);
    mix_wmma<<<waves / 8, 256, 0, stream>>>(x, mixer, xm);
  }
  // B: forward FFT, one block per frame
  fft_fwd<<<BB * CC * NF, 256, 0, stream>>>(xm, spec);
  // C: scan over frames
  {
    const int total = BB * CC * NCO;
    scan_frames<<<(total + 255) / 256, 256, 0, stream>>>(spec, transfer);
  }
  // D: inverse FFT + window + overlap-add (even store, then odd add)
  ifft_frame<<<BB * CC * (NF / 2), 256, 0, stream>>>(spec, out, 0);
  ifft_frame<<<BB * CC * (NF / 2), 256, 0, stream>>>(spec, out, 1);
  // E: tanh * gain
  {
    const size_t total = (size_t)BB * CC * TT;
    tanh_gain<<<(unsigned)((total + 255) / 256), 256, 0, stream>>>(out, gain);
  }
}